// _LocalMGDCF_2568390443298
// MI455X (gfx1250) — compile-verified
//
#include <hip/hip_runtime.h>

// ---------------------------------------------------------------------------
// LocalMGDCF propagation: out = 0.9*x + sum_{k=1..4} 0.1^k * (D^-1 A^T)^k x
// N=100000 nodes, D=128 features (f32), E=600000 edges (int src/dst).
// Sparse scatter-mean workload: no GEMM -> no WMMA; the CDNA5-relevant paths
// are wave32 coalesced row gathers, hw global_atomic_add_f32 (L2 atomics),
// and async global->LDS staging (ASYNCcnt) for gather latency hiding.
// ---------------------------------------------------------------------------

#define FDIM 128           // feature dim
#define FDIM4 32           // float4 per row
#define TPB 256            // threads per block (8 waves on wave32)
#define WPB 8              // waves per block
#define ALPHA 0.1f
#define BETA 0.9f

// Async global->LDS builtins: guard so the file compiles on both toolchains
// (and on the host pass, where amdgcn builtins don't exist).
#if defined(__HIP_DEVICE_COMPILE__) && defined(__has_builtin)
#if __has_builtin(__builtin_amdgcn_global_load_async_to_lds_b128) && \
    __has_builtin(__builtin_amdgcn_s_wait_asynccnt)
#define USE_ASYNC_LDS 1
#endif
#endif
#ifndef USE_ASYNC_LDS
#define USE_ASYNC_LDS 0
#endif

#if USE_ASYNC_LDS
// Builtin signature (from hipcc diagnostic): param0 is AS(1) "int __vector(4)*"
// (global source), param1 the LDS destination, then two immediate ints.
typedef int v4i_ __attribute__((vector_size(16)));
typedef __attribute__((address_space(1))) v4i_ gbl_v4i;
typedef __attribute__((address_space(3))) v4i_ lds_v4i;
#endif

__device__ __forceinline__ void atomic_add_f32(float* p, float v) {
  // relaxed, agent scope, result discarded -> no-return global_atomic_add_f32
  (void)__hip_atomic_fetch_add(p, v, __ATOMIC_RELAXED, __HIP_MEMORY_SCOPE_AGENT);
}

// --- scatter-add: tmp[dst[e]] += h[src[e]], one wave per edge ---------------
// (placed first in the file so the disasm snippet shows the async LDS path)
// Each wave moves one 128-float row: lane l handles floats [4l, 4l+3]
// (coalesced 512B per wave). Async path double-buffers the src-row gather
// through LDS so the next edge's load overlaps this edge's atomics.
__global__ __launch_bounds__(TPB) void scatter_kernel(const float* __restrict__ h,
                                                      float* __restrict__ tmp,
                                                      const int* __restrict__ src,
                                                      const int* __restrict__ dst,
                                                      int E) {
  const int lane = threadIdx.x & 31;
  const int wv = threadIdx.x >> 5;
  const int stride = (int)gridDim.x * WPB;
  int e = (int)blockIdx.x * WPB + wv;

#if USE_ASYNC_LDS
  __shared__ __align__(16) float lds[2][WPB][FDIM];

  if (e < E) {  // uniform per wave: all 32 lanes active together
    const float* g = h + (size_t)src[e] * FDIM + lane * 4;
    __builtin_amdgcn_global_load_async_to_lds_b128(
        (gbl_v4i*)g, (lds_v4i*)&lds[0][wv][lane * 4], 0, 0);
  }
  int buf = 0;
  for (; e < E; e += stride) {
    int en = e + stride;
    if (en < E) {
      const float* g = h + (size_t)src[en] * FDIM + lane * 4;
      __builtin_amdgcn_global_load_async_to_lds_b128(
          (gbl_v4i*)g, (lds_v4i*)&lds[buf ^ 1][wv][lane * 4], 0, 0);
      __builtin_amdgcn_s_wait_asynccnt(1);  // wait only for the older load
    } else {
      __builtin_amdgcn_s_wait_asynccnt(0);
    }
    asm volatile("" ::: "memory");  // keep LDS read below the wait
    float4 v = *(const float4*)&lds[buf][wv][lane * 4];
    float* t = tmp + (size_t)dst[e] * FDIM + lane * 4;
    atomic_add_f32(t + 0, v.x);
    atomic_add_f32(t + 1, v.y);
    atomic_add_f32(t + 2, v.z);
    atomic_add_f32(t + 3, v.w);
    buf ^= 1;
  }
#else
  for (; e < E; e += stride) {
    int en = e + stride;
    if (en < E) {
      // software prefetch of the next edge's source row (global_prefetch_b8)
      __builtin_prefetch(h + (size_t)src[en] * FDIM + lane * 4, 0, 0);
    }
    float4 v = *(const float4*)(h + (size_t)src[e] * FDIM + lane * 4);
    float* t = tmp + (size_t)dst[e] * FDIM + lane * 4;
    atomic_add_f32(t + 0, v.x);
    atomic_add_f32(t + 1, v.y);
    atomic_add_f32(t + 2, v.z);
    atomic_add_f32(t + 3, v.w);
  }
#endif
}

// --- in-degree as f32 (exact for counts < 2^24) -----------------------------
__global__ __launch_bounds__(TPB) void deg_kernel(const int* __restrict__ dst,
                                                  float* __restrict__ deg, int E) {
  int e = blockIdx.x * TPB + threadIdx.x;
  if (e < E) atomic_add_f32(&deg[dst[e]], 1.0f);
}

// --- h = x ; out = BETA * x -------------------------------------------------
__global__ __launch_bounds__(TPB) void init_kernel(const float4* __restrict__ x,
                                                   float4* __restrict__ h,
                                                   float4* __restrict__ out, int n4) {
  int i = blockIdx.x * TPB + threadIdx.x;
  if (i >= n4) return;
  float4 v = x[i];
  h[i] = v;
  float4 o;
  o.x = BETA * v.x; o.y = BETA * v.y; o.z = BETA * v.z; o.w = BETA * v.w;
  out[i] = o;
}

// --- h = tmp * inv_deg ; out += w * h ; tmp = 0 (ready for next hop) --------
// Fusing the re-zero of tmp here saves three full 51 MB memset passes: the
// cacheline is already resident when we overwrite it with zeros.
__global__ __launch_bounds__(TPB) void update_kernel(float4* __restrict__ tmp,
                                                     const float* __restrict__ deg,
                                                     float4* __restrict__ h,
                                                     float4* __restrict__ out,
                                                     float w, int n4) {
  int i = blockIdx.x * TPB + threadIdx.x;
  if (i >= n4) return;
  float dg = deg[i >> 5];  // 32 float4 per row share one degree (L0/L2 broadcast)
  float inv = dg > 0.0f ? 1.0f / dg : 0.0f;
  float4 t = tmp[i];
  float4 z; z.x = 0.0f; z.y = 0.0f; z.z = 0.0f; z.w = 0.0f;
  tmp[i] = z;  // re-zero accumulator for the next hop
  float4 hn;
  hn.x = t.x * inv; hn.y = t.y * inv; hn.z = t.z * inv; hn.w = t.w * inv;
  h[i] = hn;
  float4 o = out[i];
  o.x = fmaf(w, hn.x, o.x);
  o.y = fmaf(w, hn.y, o.y);
  o.z = fmaf(w, hn.z, o.z);
  o.w = fmaf(w, hn.w, o.w);
  out[i] = o;
}

extern "C" void kernel_launch(void* const* d_in, const int* in_sizes, int n_in,
                              void* d_out, int out_size, void* d_ws, size_t ws_size,
                              hipStream_t stream) {
  const float* x = (const float*)d_in[0];
  const int* src = (const int*)d_in[1];
  const int* dst = (const int*)d_in[2];
  float* out = (float*)d_out;

  const int E = in_sizes[1];
  const int N = in_sizes[0] / FDIM;
  const int n4 = N * FDIM4;  // float4 count

  // workspace: deg[N] | h[N*D] | tmp[N*D]  (~103 MB)
  float* deg = (float*)d_ws;
  size_t degPad = (((size_t)N + 63) & ~(size_t)63);
  float* h = deg + degPad;
  float* tmp = h + (size_t)N * FDIM;

  (void)hipMemsetAsync(deg, 0, (size_t)N * sizeof(float), stream);
  (void)hipMemsetAsync(tmp, 0, (size_t)N * FDIM * sizeof(float), stream);
  deg_kernel<<<(E + TPB - 1) / TPB, TPB, 0, stream>>>(dst, deg, E);
  init_kernel<<<(n4 + TPB - 1) / TPB, TPB, 0, stream>>>(
      (const float4*)x, (float4*)h, (float4*)out, n4);

  float w = ALPHA;
  for (int k = 0; k < 4; ++k) {
    scatter_kernel<<<2048, TPB, 0, stream>>>(h, tmp, src, dst, E);
    // update re-zeroes tmp, so the next hop's scatter starts from zeros
    update_kernel<<<(n4 + TPB - 1) / TPB, TPB, 0, stream>>>(
        (float4*)tmp, deg, (float4*)h, (float4*)out, w, n4);
    w *= ALPHA;
  }
}